// MM_STGAT_90082644067055
// MI455X (gfx1250) — compile-verified
//
#include <hip/hip_runtime.h>
#include <cmath>

// ---------------------------------------------------------------------------
// STGAT forward for MI455X (gfx1250, wave32).
// - All GEMMs: native fp32 WMMA V_WMMA_F32_16X16X4_F32 (matches fp32 reference).
// - Spatial attention: 16-wave workgroups; K/V head panels (64KB each) staged
//   into LDS once per workgroup by the Tensor Data Mover (TENSOR_LOAD_TO_LDS,
//   TENSORcnt), then a single-pass clamped/masked softmax-AV loop per wave.
//
// Input index map (top-level dict insertion order; params flattened with
// sorted keys at each level, lists in order):
//  0:x[2,12,1024,3] 1:adj[1024,1024] i32  2:auxiliary[2,10]
//  3:aux.lin.b 4:aux.lin.w 5:aux.ln.b 6:aux.ln.g
//  7:embed.lin.b 8:embed.lin.w 9:embed.ln.b 10:embed.ln.g
//  11:f1.b 12:f1.w 13:f2.b 14:f2.w 15:ga.b 16:ga.w 17:gs.b 18:gs.w
//  19:gt.b 20:gt.w 21:ln1.b 22:ln1.g 23:ln2.b 24:ln2.g
//  25:out.l1.b 26:out.l1.w 27:out.l2.b 28:out.l2.w 29:out.ln.b 30:out.ln.g
//  31..40 : spatial[0]  (k.b,k.w,ln.b,ln.g,o.b,o.w,q.b,q.w,v.b,v.w)
//  41..50 : spatial[1]
//  51..64 : temporal[0] (c1b,c1w,c2b,c2w,k.b,k.w,ln.b,ln.g,o.b,o.w,q.b,q.w,v.b,v.w)
//  65..78 : temporal[1]
// ---------------------------------------------------------------------------

#define B_  2
#define T_  12
#define N_  1024
#define H_  64
#define HD_ 16

typedef float v2f __attribute__((ext_vector_type(2)));
typedef float v8f __attribute__((ext_vector_type(8)));
typedef unsigned int u32x4 __attribute__((ext_vector_type(4)));
typedef int i32x4 __attribute__((ext_vector_type(4)));
typedef int i32x8 __attribute__((ext_vector_type(8)));

__device__ __forceinline__ v8f wmma4(v2f a, v2f b, v8f c) {
  // D = A(16x4 f32) * B(4x16 f32) + C(16x16 f32)
  return __builtin_amdgcn_wmma_f32_16x16x4_f32(false, a, false, b,
                                               (short)0, c, false, false);
}

#if __has_builtin(__builtin_amdgcn_tensor_load_to_lds)
#define HAVE_TDM 1
#else
#define HAVE_TDM 0
#endif

#if HAVE_TDM
// TDM: DMA a [1024 x 16]-f32 tile (row stride 64 floats) from global into
// LDS (rows packed contiguously, 16 floats each).  D# per CDNA5 ISA §8:
//  group0: count=1 | lds_addr | global_addr[56:0] | type=2
//  group1: data_size=4B, tensor_dim0=64, tensor_dim1=1024,
//          tile_dim0=16, tile_dim1=1024, tensor_dim0_stride=64
// This toolchain's builtin takes 6 args: (g0, g1, g2, g3, g_extra, cpol);
// groups 2/3 (and the extra group) are unused for a 2-D tensor -> zeros.
__device__ __forceinline__ void tdm_load_panel(const float* gptr, unsigned lds_off) {
  unsigned long long ga = (unsigned long long)(uintptr_t)gptr;
  u32x4 g0 = {1u,                                   // count=1 (user D#)
              lds_off,                              // LDS byte address
              (unsigned)(ga & 0xFFFFFFFFu),
              (unsigned)((ga >> 32) & 0x1FFFFFFu) | 0x80000000u};  // type=2
  i32x8 g1 = {0x00020000,                           // data_size = 4 bytes
              (int)(64u << 16),                     // tensor_dim0 = 64
              (int)(1024u << 16),                   // tensor_dim1 = 1024 (lo16)
              (int)(16u << 16),                     // tile_dim0 = 16
              1024,                                 // tile_dim1=1024, tile_dim2=0
              64,                                   // tensor_dim0_stride = 64
              0, 0};
  i32x4 z4 = {0, 0, 0, 0};
  i32x8 z8 = {0, 0, 0, 0, 0, 0, 0, 0};
  __builtin_amdgcn_tensor_load_to_lds(g0, g1, z4, z4, z8, 0);
}
#endif

// ---------------------------------------------------------------------------
// Generic Y[R,OUT] = X[R,IN] @ W[OUT,IN]^T + bias.  One wave per 16x16 tile.
// A layout: lane holds row (lane%16), K-pair {4s+2g, 4s+2g+1}, g = lane/16.
// B layout: lane holds col (lane%16), same K-pair.  (ISA 7.12.2, f32 x4.)
// ---------------------------------------------------------------------------
__global__ __launch_bounds__(32) void linear_wmma_kernel(
    const float* __restrict__ X, const float* __restrict__ W,
    const float* __restrict__ bias, float* __restrict__ Y,
    int IN, int OUT) {
  const int lane = threadIdx.x;
  const int g = lane >> 4, c16 = lane & 15;
  const int rt = blockIdx.x, ct = blockIdx.y;
  const v2f* xr = (const v2f*)(X + ((size_t)(rt * 16 + c16)) * IN + 2 * g);
  const v2f* wr = (const v2f*)(W + ((size_t)(ct * 16 + c16)) * IN + 2 * g);
  v8f acc = {};
  const int steps = IN >> 2;
  for (int s = 0; s < steps; ++s) {
    v2f a = xr[2 * s];   // floats {4s+2g, 4s+2g+1}
    v2f b = wr[2 * s];
    acc = wmma4(a, b, acc);
  }
  const float bv = bias[ct * 16 + c16];
  float* yr = Y + ((size_t)(rt * 16 + 8 * g)) * OUT + ct * 16 + c16;
#pragma unroll
  for (int vi = 0; vi < 8; ++vi)
    yr[(size_t)vi * OUT] = acc[vi] + bv;
}

// ---------------------------------------------------------------------------
// Spatial attention, single pass (scores clamped to [-10,10] => no running
// max needed; masked entries contribute exp(-10), exactly as the reference).
// Workgroup: 16 waves = 16 query tiles of one (graph m, head h).  K/V head
// panels [1024x16] are TDM-staged to LDS once per workgroup.  Per key tile:
// 4 fp32 WMMAs for S, clamp/mask/exp, P bounced through a per-wave LDS tile
// (C-layout -> A-layout), 4 more WMMAs for ctx += P*V.
// ---------------------------------------------------------------------------
__global__ __launch_bounds__(512) void spatial_attn_kernel(
    const float* __restrict__ Q, const float* __restrict__ K,
    const float* __restrict__ V, const int* __restrict__ adj,
    float* __restrict__ CTX) {
  extern __shared__ float smem[];
  float* Kp = smem;            // [1024*16] K head panel
  float* Vp = smem + 16384;    // [1024*16] V head panel
  const int tid = threadIdx.x;
  const int wave = tid >> 5, lane = tid & 31;
  const int g = lane >> 4, c16 = lane & 15;
  const int h = blockIdx.y;    // 0..3
  const int m = blockIdx.z;    // 0..23
  const int qt = blockIdx.x * 16 + wave;   // 0..63 query tile
  float* sP = smem + 32768 + wave * 256;   // per-wave 16x16 P tile

  const size_t mbase = (size_t)m * N_ * H_;

#if HAVE_TDM
  if (wave == 0) {
    tdm_load_panel(K + mbase + h * HD_, (unsigned)(uintptr_t)Kp);
    tdm_load_panel(V + mbase + h * HD_, (unsigned)(uintptr_t)Vp);
    __builtin_amdgcn_s_wait_tensorcnt(0);
  }
#else
  for (int i = tid; i < N_ * HD_; i += 512) {
    const int n = i >> 4, d = i & 15;
    Kp[i] = K[mbase + (size_t)n * H_ + h * HD_ + d];
    Vp[i] = V[mbase + (size_t)n * H_ + h * HD_ + d];
  }
#endif
  __syncthreads();

  // Preload Q tile in A layout, pre-scaled by 1/sqrt(HD)=0.25 (scale-then-clip).
  const float* qr = Q + mbase + (size_t)(qt * 16 + c16) * H_ + h * HD_ + 2 * g;
  v2f aq[4];
#pragma unroll
  for (int s = 0; s < 4; ++s)
    aq[s] = (v2f){qr[4 * s] * 0.25f, qr[4 * s + 1] * 0.25f};

  v8f acc = {};
  float rs[8] = {0.f, 0.f, 0.f, 0.f, 0.f, 0.f, 0.f, 0.f};

  for (int kt = 0; kt < 64; ++kt) {
    // S = (Q/4) * K^T   (B-matrix cols = key nodes; K from LDS panel)
    const float* kr = Kp + (kt * 16 + c16) * HD_ + 2 * g;
    v8f sv = {};
#pragma unroll
    for (int s = 0; s < 4; ++s) {
      v2f b = (v2f){kr[4 * s], kr[4 * s + 1]};
      sv = wmma4(aq[s], b, sv);
    }
    // clamp, adjacency mask (-> exactly -10), exp; stash P in per-wave LDS.
    const int* arow = adj + (size_t)(qt * 16 + 8 * g) * N_ + kt * 16 + c16;
#pragma unroll
    for (int vi = 0; vi < 8; ++vi) {
      float xv = sv[vi];
      xv = fminf(fmaxf(xv, -10.0f), 10.0f);
      int a = arow[(size_t)vi * N_];
      xv = (a == 0) ? -10.0f : xv;
      float e = __expf(xv);
      rs[vi] += e;
      sP[(vi + 8 * g) * 16 + c16] = e;
    }
    // ctx += P * V  (P re-read in A layout; per-wave LDS ops are in-order,
    // no workgroup barrier needed; V from LDS panel in B layout)
    const float* vr = Vp + (kt * 16) * HD_ + c16;
#pragma unroll
    for (int s = 0; s < 4; ++s) {
      const int kk = 4 * s + 2 * g;
      v2f a = (v2f){sP[c16 * 16 + kk], sP[c16 * 16 + kk + 1]};
      v2f b = (v2f){vr[(size_t)kk * HD_], vr[(size_t)(kk + 1) * HD_]};
      acc = wmma4(a, b, acc);
    }
  }
  // Row sums: reduce within each 16-lane group (C/D row = vi + 8*g).
#pragma unroll
  for (int vi = 0; vi < 8; ++vi) {
    float r = rs[vi];
    r += __shfl_xor(r, 1, 32);
    r += __shfl_xor(r, 2, 32);
    r += __shfl_xor(r, 4, 32);
    r += __shfl_xor(r, 8, 32);
    rs[vi] = r;
  }
  float* cr = CTX + mbase + (size_t)(qt * 16 + 8 * g) * H_ + h * HD_ + c16;
#pragma unroll
  for (int vi = 0; vi < 8; ++vi)
    cr[(size_t)vi * H_] = acc[vi] / rs[vi];
}

// ---------------------------------------------------------------------------
// Row LayerNorm (dim = blockDim.x in {32,64,128}), optional residual + relu.
// ---------------------------------------------------------------------------
__global__ void ln_kernel(const float* __restrict__ X, const float* __restrict__ RES,
                          const float* __restrict__ Gm, const float* __restrict__ Bt,
                          float* __restrict__ Y, int do_relu) {
  const int dim = blockDim.x;
  const int j = threadIdx.x;
  const size_t row = blockIdx.x;
  __shared__ float red[128];
  float v = X[row * dim + j];
  if (RES) v += RES[row * dim + j];
  red[j] = v;
  __syncthreads();
  for (int s = dim >> 1; s > 0; s >>= 1) {
    if (j < s) red[j] += red[j + s];
    __syncthreads();
  }
  const float mean = red[0] / (float)dim;
  __syncthreads();
  const float d = v - mean;
  red[j] = d * d;
  __syncthreads();
  for (int s = dim >> 1; s > 0; s >>= 1) {
    if (j < s) red[j] += red[j + s];
    __syncthreads();
  }
  const float var = red[0] / (float)dim;
  float y = d * rsqrtf(var + 1e-5f) * Gm[j] + Bt[j];
  if (do_relu) y = fmaxf(y, 0.0f);
  Y[row * dim + j] = y;
}

// Embed: y = relu(LN(x[.,3] @ W[64,3]^T + b)) ; one 64-thread block per row.
__global__ void embed_kernel(const float* __restrict__ X, const float* __restrict__ W,
                             const float* __restrict__ bias, const float* __restrict__ Gm,
                             const float* __restrict__ Bt, float* __restrict__ Y) {
  const size_t row = blockIdx.x;
  const int j = threadIdx.x;  // 64
  __shared__ float red[64];
  const float x0 = X[row * 3], x1 = X[row * 3 + 1], x2 = X[row * 3 + 2];
  float v = W[j * 3] * x0 + W[j * 3 + 1] * x1 + W[j * 3 + 2] * x2 + bias[j];
  red[j] = v;
  __syncthreads();
  for (int s = 32; s > 0; s >>= 1) {
    if (j < s) red[j] += red[j + s];
    __syncthreads();
  }
  const float mean = red[0] * (1.0f / 64.0f);
  __syncthreads();
  const float d = v - mean;
  red[j] = d * d;
  __syncthreads();
  for (int s = 32; s > 0; s >>= 1) {
    if (j < s) red[j] += red[j + s];
    __syncthreads();
  }
  const float var = red[0] * (1.0f / 64.0f);
  Y[row * 64 + j] = fmaxf(d * rsqrtf(var + 1e-5f) * Gm[j] + Bt[j], 0.0f);
}

// Conv1d k=3 'same' over [M,T,Cin] -> [M,T,Cout], exact GELU.
__global__ void conv_gelu_kernel(const float* __restrict__ X, const float* __restrict__ W,
                                 const float* __restrict__ bias, float* __restrict__ Y,
                                 int Cin, int Cout) {
  const int mt = blockIdx.x;  // m*T + t
  const int m = mt / T_, t = mt % T_;
  const int o = threadIdx.x;  // Cout
  float acc = bias[o];
  const float* wrow = W + (size_t)o * Cin * 3;  // w[o][i][dk]
  for (int dk = 0; dk < 3; ++dk) {
    const int tt = t + dk - 1;
    if (tt < 0 || tt >= T_) continue;
    const float* xr = X + ((size_t)m * T_ + tt) * Cin;
    for (int i = 0; i < Cin; ++i) acc += xr[i] * wrow[i * 3 + dk];
  }
  Y[(size_t)mt * Cout + o] = 0.5f * acc * (1.0f + erff(acc * 0.70710678118654752f));
}

// Temporal attention: T=12, per-thread (m, t, h) — tiny, VALU.
__global__ void temporal_attn_kernel(const float* __restrict__ Q, const float* __restrict__ K,
                                     const float* __restrict__ V, float* __restrict__ CTX) {
  const int m = blockIdx.x;   // 0..2047
  const int t = threadIdx.x;  // 0..11
  const int h = threadIdx.y;  // 0..3
  const float* q = Q + ((size_t)m * T_ + t) * H_ + h * HD_;
  float qv[HD_];
#pragma unroll
  for (int d = 0; d < HD_; ++d) qv[d] = q[d];
  float sc[T_];
  float mx = -1e30f;
  for (int kk = 0; kk < T_; ++kk) {
    const float* kr = K + ((size_t)m * T_ + kk) * H_ + h * HD_;
    float s = 0.f;
#pragma unroll
    for (int d = 0; d < HD_; ++d) s += qv[d] * kr[d];
    s *= 0.25f;
    sc[kk] = s;
    mx = fmaxf(mx, s);
  }
  float den = 0.f;
  for (int kk = 0; kk < T_; ++kk) { sc[kk] = __expf(sc[kk] - mx); den += sc[kk]; }
  const float inv = 1.0f / den;
  float ov[HD_];
#pragma unroll
  for (int d = 0; d < HD_; ++d) ov[d] = 0.f;
  for (int kk = 0; kk < T_; ++kk) {
    const float* vr = V + ((size_t)m * T_ + kk) * H_ + h * HD_;
    const float a = sc[kk] * inv;
#pragma unroll
    for (int d = 0; d < HD_; ++d) ov[d] += a * vr[d];
  }
  float* c = CTX + ((size_t)m * T_ + t) * H_ + h * HD_;
#pragma unroll
  for (int d = 0; d < HD_; ++d) c[d] = ov[d];
}

// spatial_feat[b,n,:] = mean over T of hs[b,t,n,:]
__global__ void mean_t_kernel(const float* __restrict__ HS, float* __restrict__ SF) {
  const int row = blockIdx.x;  // b*N + n
  const int b = row >> 10, n = row & (N_ - 1);
  const int j = threadIdx.x;   // 64
  float acc = 0.f;
  for (int t = 0; t < T_; ++t)
    acc += HS[(((size_t)b * T_ + t) * N_ + n) * H_ + j];
  SF[(size_t)row * H_ + j] = acc * (1.0f / 12.0f);
}

// hs[B,T,N,H] -> ht[B,N,T,H]
__global__ void transpose_kernel(const float* __restrict__ HS, float* __restrict__ HT) {
  const size_t row = blockIdx.x;  // over B*N*T
  const int j = threadIdx.x;      // 64
  const int b = (int)(row / ((size_t)N_ * T_));
  const int rem = (int)(row % ((size_t)N_ * T_));
  const int n = rem / T_, t = rem % T_;
  HT[row * H_ + j] = HS[(((size_t)b * T_ + t) * N_ + n) * H_ + j];
}

// aux: relu(LN(aux[b,10] @ W[32,10]^T + b)) ; one wave per batch row.
__global__ __launch_bounds__(32) void aux_kernel(
    const float* __restrict__ A, const float* __restrict__ W,
    const float* __restrict__ bias, const float* __restrict__ Gm,
    const float* __restrict__ Bt, float* __restrict__ Y) {
  const int b = blockIdx.x, j = threadIdx.x;  // 32
  float acc = bias[j];
  for (int i = 0; i < 10; ++i) acc += A[b * 10 + i] * W[j * 10 + i];
  float s = acc;
  for (int o = 16; o > 0; o >>= 1) s += __shfl_xor(s, o, 32);
  const float mean = s * (1.0f / 32.0f);
  const float d = acc - mean;
  float s2 = d * d;
  for (int o = 16; o > 0; o >>= 1) s2 += __shfl_xor(s2, o, 32);
  const float var = s2 * (1.0f / 32.0f);
  Y[b * 32 + j] = fmaxf(d * rsqrtf(var + 1e-5f) * Gm[j] + Bt[j], 0.0f);
}

// Gates + gated concat: comb[row,160] ; one wave per (b,n).
__global__ __launch_bounds__(32) void fusion_gate_kernel(
    const float* __restrict__ SF, const float* __restrict__ HT, const float* __restrict__ AF,
    const float* __restrict__ wgs, const float* __restrict__ bgs,
    const float* __restrict__ wgt, const float* __restrict__ bgt,
    const float* __restrict__ wga, const float* __restrict__ bga,
    float* __restrict__ CB) {
  const int row = blockIdx.x;  // b*N + n
  const int b = row >> 10;
  const int lane = threadIdx.x;
  const float* sf = SF + (size_t)row * H_;
  const float* tf = HT + ((size_t)row * T_ + (T_ - 1)) * H_;  // ht[:, -1, :]
  const float* af = AF + b * 32;
  float ds = 0.f, dt = 0.f;
  for (int j = lane; j < H_; j += 32) { ds += sf[j] * wgs[j]; dt += tf[j] * wgt[j]; }
  float da = af[lane] * wga[lane];
  for (int o = 16; o > 0; o >>= 1) {
    ds += __shfl_xor(ds, o, 32);
    dt += __shfl_xor(dt, o, 32);
    da += __shfl_xor(da, o, 32);
  }
  const float gs = 1.0f / (1.0f + __expf(-(ds + bgs[0])));
  const float gt = 1.0f / (1.0f + __expf(-(dt + bgt[0])));
  const float ga = 1.0f / (1.0f + __expf(-(da + bga[0])));
  const float inv = 1.0f / (gs + gt + ga);
  const float cs = gs * inv, ct = gt * inv, ca = ga * inv;
  float* out = CB + (size_t)row * 160;
  for (int j = lane; j < H_; j += 32) { out[j] = sf[j] * cs; out[64 + j] = tf[j] * ct; }
  out[128 + lane] = af[lane] * ca;
}

// Final head: out[row,12] = O1[row,32] @ W[12,32]^T + b
__global__ void out_l2_kernel(const float* __restrict__ X, const float* __restrict__ W,
                              const float* __restrict__ bias, float* __restrict__ Y) {
  const int row = blockIdx.x;
  const int j = threadIdx.x;  // 16, use 12
  if (j < 12) {
    float acc = bias[j];
    const float* xr = X + (size_t)row * 32;
    const float* wr = W + j * 32;
    for (int i = 0; i < 32; ++i) acc += xr[i] * wr[i];
    Y[(size_t)row * 12 + j] = acc;
  }
}

// ---------------------------------------------------------------------------
extern "C" void kernel_launch(void* const* d_in, const int* in_sizes, int n_in,
                              void* d_out, int out_size, void* d_ws, size_t ws_size,
                              hipStream_t stream) {
  (void)in_sizes; (void)n_in; (void)out_size; (void)ws_size;
  auto F = [&](int i) { return (const float*)d_in[i]; };
  const float* x = F(0);
  const int* adj = (const int*)d_in[1];
  const float* auxi = F(2);

  float* ws = (float*)d_ws;
  const size_t S = (size_t)24576 * 64;  // one [B*T*N, H] activation slab
  float *HS = ws,        *Qb = ws + S,     *Kb = ws + 2 * S, *Vb = ws + 3 * S;
  float *CT = ws + 4 * S, *T1 = ws + 5 * S, *HT = ws + 6 * S;
  float *C1 = ws + 7 * S;                 // [2048,12,128] = 2 slabs
  float *C2 = ws + 9 * S;                 // [2048,12,128] = 2 slabs
  float *SF = ws + 11 * S;                // [2048,64]
  float *AF = SF + (size_t)2048 * 64;     // [2,32]
  float *CB = AF + 64;                    // [2048,160]
  float *F1b = CB + (size_t)2048 * 160;   // [2048,128]
  float *F2b = F1b + (size_t)2048 * 128;  // [2048,64]
  float *O1 = F2b + (size_t)2048 * 64;    // [2048,32]

  const size_t attn_lds = (size_t)(32768 + 16 * 256) * sizeof(float);  // 144 KB

  // ---- embed: relu(LN(x @ W^T + b)) -> HS [24576,64]
  embed_kernel<<<24576, 64, 0, stream>>>(x, F(8), F(7), F(10), F(9), HS);

  // ---- spatial layers (params: k.b,k.w,ln.b,ln.g,o.b,o.w,q.b,q.w,v.b,v.w)
  for (int L = 0; L < 2; ++L) {
    const int p = 31 + 10 * L;
    linear_wmma_kernel<<<dim3(1536, 4), 32, 0, stream>>>(HS, F(p + 7), F(p + 6), Qb, 64, 64);
    linear_wmma_kernel<<<dim3(1536, 4), 32, 0, stream>>>(HS, F(p + 1), F(p + 0), Kb, 64, 64);
    linear_wmma_kernel<<<dim3(1536, 4), 32, 0, stream>>>(HS, F(p + 9), F(p + 8), Vb, 64, 64);
    spatial_attn_kernel<<<dim3(4, 4, 24), 512, attn_lds, stream>>>(Qb, Kb, Vb, adj, CT);
    linear_wmma_kernel<<<dim3(1536, 4), 32, 0, stream>>>(CT, F(p + 5), F(p + 4), T1, 64, 64);
    ln_kernel<<<24576, 64, 0, stream>>>(T1, HS, F(p + 3), F(p + 2), HS, 0);
  }

  // ---- spatial_feat (mean over T), then hs -> ht transpose
  mean_t_kernel<<<2048, 64, 0, stream>>>(HS, SF);
  transpose_kernel<<<24576, 64, 0, stream>>>(HS, HT);

  // ---- temporal layers (c1b,c1w,c2b,c2w,k.b,k.w,ln.b,ln.g,o.b,o.w,q.b,q.w,v.b,v.w)
  for (int L = 0; L < 2; ++L) {
    const int p = 51 + 14 * L;
    conv_gelu_kernel<<<24576, 128, 0, stream>>>(HT, F(p + 1), F(p + 0), C1, 64, 128);
    conv_gelu_kernel<<<24576, 128, 0, stream>>>(C1, F(p + 3), F(p + 2), C2, 128, 128);
    linear_wmma_kernel<<<dim3(1536, 4), 32, 0, stream>>>(C2, F(p + 11), F(p + 10), Qb, 128, 64);
    linear_wmma_kernel<<<dim3(1536, 4), 32, 0, stream>>>(C2, F(p + 5),  F(p + 4),  Kb, 128, 64);
    linear_wmma_kernel<<<dim3(1536, 4), 32, 0, stream>>>(C2, F(p + 13), F(p + 12), Vb, 128, 64);
    temporal_attn_kernel<<<2048, dim3(12, 4), 0, stream>>>(Qb, Kb, Vb, CT);
    linear_wmma_kernel<<<dim3(1536, 4), 32, 0, stream>>>(CT, F(p + 9), F(p + 8), T1, 64, 64);
    ln_kernel<<<24576, 64, 0, stream>>>(T1, HT, F(p + 7), F(p + 6), HT, 0);
  }

  // ---- aux branch, gates + gated concat
  aux_kernel<<<2, 32, 0, stream>>>(auxi, F(4), F(3), F(6), F(5), AF);
  fusion_gate_kernel<<<2048, 32, 0, stream>>>(SF, HT, AF,
                                              F(18), F(17),   // gs.w, gs.b
                                              F(20), F(19),   // gt.w, gt.b
                                              F(16), F(15),   // ga.w, ga.b
                                              CB);

  // ---- fusion MLP + output head
  linear_wmma_kernel<<<dim3(128, 8), 32, 0, stream>>>(CB, F(12), F(11), F1b, 160, 128);
  ln_kernel<<<2048, 128, 0, stream>>>(F1b, nullptr, F(22), F(21), F1b, 1);
  linear_wmma_kernel<<<dim3(128, 4), 32, 0, stream>>>(F1b, F(14), F(13), F2b, 128, 64);
  ln_kernel<<<2048, 64, 0, stream>>>(F2b, nullptr, F(24), F(23), F2b, 1);
  linear_wmma_kernel<<<dim3(128, 2), 32, 0, stream>>>(F2b, F(26), F(25), O1, 64, 32);
  ln_kernel<<<2048, 32, 0, stream>>>(O1, nullptr, F(30), F(29), O1, 1);
  out_l2_kernel<<<2048, 16, 0, stream>>>(O1, F(28), F(27), (float*)d_out);
}